// Involution_7593502179575
// MI455X (gfx1250) — compile-verified
//
#include <hip/hip_runtime.h>

typedef __attribute__((ext_vector_type(2))) float v2f;
typedef __attribute__((ext_vector_type(8))) float v8f;

#define EPS 1e-5f
#define HW  3136      // 56*56
#define CIN 256
#define CR  64
#define GK  784       // 16 groups * 49 taps
#define NB  4

// -------------------------------------------------------------------------
// Kernel 1: t[b,o,p] = relu(BN(sum_c W1[o,c] * x[b,c,p] + b1[o]))
// One wave computes a 16(o) x 16(pixel) tile via V_WMMA_F32_16X16X4_F32,
// looping K over 256 channels in steps of 4. Block = 128 threads = 4 waves
// covering all 64 output channels for one 16-pixel strip.
// -------------------------------------------------------------------------
__global__ void __launch_bounds__(128)
invo_gemm1_bn_relu(const float* __restrict__ x, const float* __restrict__ W1,
                   const float* __restrict__ b1, const float* __restrict__ gamma,
                   const float* __restrict__ beta, const float* __restrict__ mean,
                   const float* __restrict__ var, float* __restrict__ t)
{
    const int lane   = threadIdx.x & 31;
    const int wave   = threadIdx.x >> 5;      // o-tile 0..3
    const int laneN  = lane & 15;             // N (pixel) index inside tile
    const int laneHi = lane >> 4;             // 0: K=0,1  1: K=2,3 (A/B frag halves)
    const int p0     = blockIdx.x * 16;       // pixel strip
    const int b      = blockIdx.y;            // batch
    const int o0     = wave * 16;             // output-channel tile

    const float* xb   = x + (size_t)b * CIN * HW;
    // A fragment source: W1[o0+laneN][k + 2*laneHi + {0,1}] -> one b64 load/lane
    const float* Arow = W1 + (size_t)(o0 + laneN) * CIN + 2 * laneHi;
    // B fragment source: x[b][k + 2*laneHi + r][p0 + laneN], channel stride HW
    const float* Bcol = xb + (size_t)(2 * laneHi) * HW + p0 + laneN;

    v8f acc = {};
#pragma unroll 8
    for (int k = 0; k < CIN; k += 4) {
        v2f a = *(const v2f*)(Arow + k);                 // K = k+2h, k+2h+1
        v2f bb;
        bb.x = Bcol[(size_t)k * HW];                     // K = k+2h
        bb.y = Bcol[(size_t)(k + 1) * HW];               // K = k+2h+1
        acc = __builtin_amdgcn_wmma_f32_16x16x4_f32(
            false, a, false, bb, (short)0, acc, false, false);
    }

    // Epilogue: D element r lives at (M = r + 8*laneHi, N = laneN)
    float* tb = t + (size_t)b * CR * HW + p0 + laneN;
#pragma unroll
    for (int r = 0; r < 8; ++r) {
        int o = o0 + r + 8 * laneHi;
        float s  = gamma[o] * __frsqrt_rn(var[o] + EPS);
        float v  = (acc[r] + b1[o]) * s + (beta[o] - mean[o] * s);
        tb[(size_t)o * HW] = fmaxf(v, 0.0f);
    }
}

// -------------------------------------------------------------------------
// Kernel 2: w[b,m,p] = sum_c W2[m,c] * t[b,c,p] + b2[m]   (m in [0,784))
// One wave per block: 16(m) x 16(pixel) tile, K loop over 64 in steps of 4.
// -------------------------------------------------------------------------
__global__ void __launch_bounds__(32)
invo_gemm2(const float* __restrict__ t, const float* __restrict__ W2,
           const float* __restrict__ b2, float* __restrict__ w)
{
    const int lane   = threadIdx.x & 31;
    const int laneN  = lane & 15;
    const int laneHi = lane >> 4;
    const int p0     = blockIdx.x * 16;       // 0..195 strips
    const int m0     = blockIdx.y * 16;       // 0..48 m-tiles (784/16)
    const int b      = blockIdx.z;

    const float* tb   = t + (size_t)b * CR * HW;
    const float* Arow = W2 + (size_t)(m0 + laneN) * CR + 2 * laneHi;
    const float* Bcol = tb + (size_t)(2 * laneHi) * HW + p0 + laneN;

    v8f acc = {};
#pragma unroll
    for (int k = 0; k < CR; k += 4) {
        v2f a = *(const v2f*)(Arow + k);
        v2f bb;
        bb.x = Bcol[(size_t)k * HW];
        bb.y = Bcol[(size_t)(k + 1) * HW];
        acc = __builtin_amdgcn_wmma_f32_16x16x4_f32(
            false, a, false, bb, (short)0, acc, false, false);
    }

    float* wb = w + (size_t)b * GK * HW + p0 + laneN;
#pragma unroll
    for (int r = 0; r < 8; ++r) {
        int m = m0 + r + 8 * laneHi;
        wb[(size_t)m * HW] = acc[r] + b2[m];
    }
}

// -------------------------------------------------------------------------
// Kernel 3: involution. Block = 256 threads = 16 group-channels x 16 pixels,
// handles one (batch, group, 16-pixel strip). The 49x16 per-pixel weights are
// staged once in LDS and reused by all 16 channels of the group.
// -------------------------------------------------------------------------
__global__ void __launch_bounds__(256)
invo_apply(const float* __restrict__ x, const float* __restrict__ wk,
           float* __restrict__ out)
{
    __shared__ float wl[49 * 16];

    const int p0 = blockIdx.x * 16;   // pixel strip
    const int g  = blockIdx.y;        // group 0..15
    const int b  = blockIdx.z;        // batch
    const int tid = threadIdx.x;

    // cooperative load of 49 taps x 16 pixels into LDS
    const float* wsrc = wk + ((size_t)b * GK + (size_t)g * 49) * HW + p0;
#pragma unroll
    for (int idx = tid; idx < 49 * 16; idx += 256) {
        int kk = idx >> 4, pl = idx & 15;
        wl[kk * 16 + pl] = wsrc[(size_t)kk * HW + pl];
    }
    __syncthreads();

    const int pl = tid & 15;          // pixel in strip
    const int cg = tid >> 4;          // channel within group
    const int p  = p0 + pl;
    const int h  = p / 56;
    const int wc = p % 56;
    const int c  = g * 16 + cg;

    const float* xc = x + ((size_t)b * CIN + c) * HW;

    float acc = 0.0f;
#pragma unroll
    for (int i = 0; i < 7; ++i) {
        int hh = h + i - 3;
        bool hin = (hh >= 0) & (hh < 56);
#pragma unroll
        for (int j = 0; j < 7; ++j) {
            int ww = wc + j - 3;
            bool ok = hin & (ww >= 0) & (ww < 56);
            float xv = ok ? xc[hh * 56 + ww] : 0.0f;
            acc = fmaf(xv, wl[(i * 7 + j) * 16 + pl], acc);
        }
    }
    out[((size_t)b * CIN + c) * HW + p] = acc;
}

// -------------------------------------------------------------------------
extern "C" void kernel_launch(void* const* d_in, const int* in_sizes, int n_in,
                              void* d_out, int out_size, void* d_ws, size_t ws_size,
                              hipStream_t stream)
{
    const float* x     = (const float*)d_in[0];
    const float* W1    = (const float*)d_in[1];
    const float* b1    = (const float*)d_in[2];
    const float* gamma = (const float*)d_in[3];
    const float* beta  = (const float*)d_in[4];
    const float* mean  = (const float*)d_in[5];
    const float* var   = (const float*)d_in[6];
    const float* W2    = (const float*)d_in[7];
    const float* b2    = (const float*)d_in[8];
    float* out = (float*)d_out;

    float* t = (float*)d_ws;                                   // 4*64*3136 f32
    float* w = t + (size_t)NB * CR * HW;                       // 4*784*3136 f32

    // GEMM1 + BN + ReLU: grid (pixel strips, batch), 4 waves cover 64 out-ch
    dim3 g1(HW / 16, NB);
    invo_gemm1_bn_relu<<<g1, 128, 0, stream>>>(x, W1, b1, gamma, beta, mean, var, t);

    // GEMM2: grid (pixel strips, 49 m-tiles, batch), 1 wave per block
    dim3 g2(HW / 16, GK / 16, NB);
    invo_gemm2<<<g2, 32, 0, stream>>>(t, W2, b2, w);

    // Involution: grid (pixel strips, groups, batch)
    dim3 g3(HW / 16, 16, NB);
    invo_apply<<<g3, 256, 0, stream>>>(x, w, out);
}